// AdaptiveNet_67482526155268
// MI455X (gfx1250) — compile-verified
//
#include <hip/hip_runtime.h>
#include <stdint.h>

// AdaptiveNet on MI455X (gfx1250, wave32).
//   K1: fc1 + sigmoid + grouped-linear fused -> x2 (workspace)
//   K2: fc3
// fp32 WMMA (16x16x4) keeps full precision; the workload is HBM-bound
// (~130 MB min traffic @ 23.3 TB/s), so data movement is what we optimize:
// async global->LDS DMA (ASYNCcnt) + double buffering + pipelined ds_loads.

typedef float v2f __attribute__((ext_vector_type(2)));
typedef float v8f __attribute__((ext_vector_type(8)));
typedef int   v4i __attribute__((vector_size(16)));   // matches builtin pointee

typedef __attribute__((address_space(1))) v4i* gptr4;  // global int4*
typedef __attribute__((address_space(3))) v4i* lptr4;  // LDS int4*

#define B_SZ   16384
#define D_IN   1024
#define H1     4096
#define H2     512
#define D_OUT  256

#define KSTEP  32
#define LDSS   36                 // padded row stride (floats): 144B rows, 16B aligned
#define TILEF  (64 * LDSS)        // one 64x32 tile in LDS (floats)
#define STAGEF (2 * TILEF)        // A tile + B tile per stage

#if defined(__AMDGCN__) && __has_builtin(__builtin_amdgcn_global_load_async_to_lds_b128)
#define ASYNC_LDS 1
#else
#define ASYNC_LDS 0
#endif

__device__ __forceinline__ void async_copy16(const float* g, float* l) {
#if ASYNC_LDS
  // flat LDS addresses carry the LDS byte offset in the low 32 bits (ISA 10.2);
  // flat global addresses are identity -> integer casts give valid AS1/AS3 ptrs.
  __builtin_amdgcn_global_load_async_to_lds_b128(
      (gptr4)(uintptr_t)g,
      (lptr4)(uint32_t)(uintptr_t)l,
      0, 0);
#else
  *(float4*)l = *(const float4*)g;
#endif
}

template <int N>
__device__ __forceinline__ void wait_async_lds() {
#if ASYNC_LDS
#if __has_builtin(__builtin_amdgcn_s_wait_asynccnt)
  __builtin_amdgcn_s_wait_asynccnt(N);
#else
  asm volatile("s_wait_asynccnt %0" ::"n"(N));
#endif
#endif
}

__device__ __forceinline__ v8f wmma_f32_k4(v2f a, v2f b, v8f c) {
  // (neg_a, A, neg_b, B, c_mod, C, reuse_a, reuse_b)
  return __builtin_amdgcn_wmma_f32_16x16x4_f32(false, a, false, b, (short)0, c,
                                               false, false);
}

__device__ __forceinline__ float sigmoidf_fast(float v) {
  return 1.0f / (1.0f + __expf(-v));
}

// ---------------------------------------------------------------------------
// Shared 64(M)x64(N) WMMA mainloop, K-staged through double-buffered LDS via
// async DMA. Both operands are row-major with row length == K (x/W1, x2/W3).
// Block: 256 threads = 8 waves as 2(M) x 4(N); 2 accumulators per wave.
// ---------------------------------------------------------------------------
__device__ __forceinline__ void wmma_mainloop(
    const float* __restrict__ Ag, const float* __restrict__ Bg, int K, int m0,
    int n0, float* smem, v8f& acc0, v8f& acc1) {
  const int t    = threadIdx.x;
  const int lane = t & 31;
  const int wave = t >> 5;
  const int wr   = wave >> 2;          // 0..1 : M half
  const int wc   = wave & 3;           // 0..3 : N quarter
  const int l15  = lane & 15;
  const int krow = (lane >> 4) * 2;    // K-pair base of the f32 WMMA A/B layout

  auto issue = [&](int s, int buf) {
    const int k0 = s * KSTEP;
    float* la = smem + buf * STAGEF;
    float* lb = la + TILEF;
#pragma unroll
    for (int i = 0; i < 2; ++i) {      // 64x32 tile = 2048 floats = 256thr*2*f4
      const int e = (i * 256 + t) * 4;
      const int r = e >> 5, c = e & 31;
      async_copy16(&Ag[(size_t)(m0 + r) * K + k0 + c], &la[r * LDSS + c]);
      async_copy16(&Bg[(size_t)(n0 + r) * K + k0 + c], &lb[r * LDSS + c]);
    }
  };

  const int nstage = K / KSTEP;
  issue(0, 0);
  for (int s = 0; s < nstage; ++s) {
    const int buf = s & 1;
    if (s + 1 < nstage) {
      issue(s + 1, buf ^ 1);           // 4 more asyncs outstanding
      wait_async_lds<4>();             // stage s's 4 have landed (in-order)
    } else {
      wait_async_lds<0>();
    }
    __syncthreads();

    const float* la = smem + buf * STAGEF;
    const float* lb = la + TILEF;

    // one-step software pipeline on fragments: ds_loads for k+4 in flight
    // while the wmmas for k execute.
    v2f a0c = *(const v2f*)(&la[(wr * 32 +      l15) * LDSS + krow]);
    v2f a1c = *(const v2f*)(&la[(wr * 32 + 16 + l15) * LDSS + krow]);
    v2f bc  = *(const v2f*)(&lb[(wc * 16 +      l15) * LDSS + krow]);
#pragma unroll
    for (int k = 0; k < KSTEP; k += 4) {
      v2f a0n, a1n, bn;
      if (k + 4 < KSTEP) {
        const int kk = k + 4 + krow;
        a0n = *(const v2f*)(&la[(wr * 32 +      l15) * LDSS + kk]);
        a1n = *(const v2f*)(&la[(wr * 32 + 16 + l15) * LDSS + kk]);
        bn  = *(const v2f*)(&lb[(wc * 16 +      l15) * LDSS + kk]);
      }
      acc0 = wmma_f32_k4(a0c, bc, acc0);
      acc1 = wmma_f32_k4(a1c, bc, acc1);
      a0c = a0n; a1c = a1n; bc = bn;
    }
    __syncthreads();
  }
}

// ---------------------------------------------------------------------------
// K1: x2[b,g] = sigmoid( sum_s sigmoid(x@W1.T + b1)[b,g*8+s]*W2[g,s] + b2[g] )
// W2 is [512,8] row-major -> W2[g,s] == W2flat[n] with n = g*8+s, so the
// grouped layer is an elementwise scale + 8-lane butterfly reduce.
// ---------------------------------------------------------------------------
__global__ __launch_bounds__(256) void fc1_group_fused(
    const float* __restrict__ x,  const float* __restrict__ W1,
    const float* __restrict__ b1, const float* __restrict__ W2,
    const float* __restrict__ b2, float* __restrict__ x2) {
  __shared__ float smem[2 * STAGEF];

  const int t    = threadIdx.x;
  const int lane = t & 31;
  const int wave = t >> 5;
  const int wr   = wave >> 2;
  const int wc   = wave & 3;
  const int l15  = lane & 15;
  const int half = lane >> 4;
  const int m0   = blockIdx.x * 64;
  const int n0   = blockIdx.y * 64;

  v8f acc0 = {};
  v8f acc1 = {};
  wmma_mainloop(x, W1, D_IN, m0, n0, smem, acc0, acc1);

  const int   ng  = n0 + wc * 16 + l15;   // global H1 column
  const float b1v = b1[ng];
  const float w2v = W2[ng];

#pragma unroll
  for (int tt = 0; tt < 2; ++tt) {
    v8f acc = tt ? acc1 : acc0;
#pragma unroll
    for (int j = 0; j < 8; ++j) {
      float h = sigmoidf_fast(acc[j] + b1v);
      float p = h * w2v;
      p += __shfl_xor(p, 1, 32);
      p += __shfl_xor(p, 2, 32);
      p += __shfl_xor(p, 4, 32);
      if ((lane & 7) == 0) {
        const int row = m0 + wr * 32 + tt * 16 + j + half * 8;
        const int g   = (n0 >> 3) + wc * 2 + (l15 >> 3);
        x2[(size_t)row * H2 + g] = sigmoidf_fast(p + b2[g]);
      }
    }
  }
}

// ---------------------------------------------------------------------------
// K2: out = x2 @ W3.T + b3   ([16384,512] x [512,256])
// ---------------------------------------------------------------------------
__global__ __launch_bounds__(256) void fc3_kernel(
    const float* __restrict__ x2, const float* __restrict__ W3,
    const float* __restrict__ b3, float* __restrict__ out) {
  __shared__ float smem[2 * STAGEF];

  const int t    = threadIdx.x;
  const int lane = t & 31;
  const int wave = t >> 5;
  const int wr   = wave >> 2;
  const int wc   = wave & 3;
  const int l15  = lane & 15;
  const int half = lane >> 4;
  const int m0   = blockIdx.x * 64;
  const int n0   = blockIdx.y * 64;

  v8f acc0 = {};
  v8f acc1 = {};
  wmma_mainloop(x2, W3, H2, m0, n0, smem, acc0, acc1);

  const int   ng  = n0 + wc * 16 + l15;   // global D_OUT column
  const float b3v = b3[ng];

#pragma unroll
  for (int tt = 0; tt < 2; ++tt) {
    v8f acc = tt ? acc1 : acc0;
#pragma unroll
    for (int j = 0; j < 8; ++j) {
      const int row = m0 + wr * 32 + tt * 16 + j + half * 8;
      out[(size_t)row * D_OUT + ng] = acc[j] + b3v;
    }
  }
}

// ---------------------------------------------------------------------------
extern "C" void kernel_launch(void* const* d_in, const int* in_sizes, int n_in,
                              void* d_out, int out_size, void* d_ws, size_t ws_size,
                              hipStream_t stream) {
  (void)in_sizes; (void)n_in; (void)out_size; (void)ws_size;

  const float* x  = (const float*)d_in[0];
  const float* W1 = (const float*)d_in[1];
  const float* b1 = (const float*)d_in[2];
  const float* W2 = (const float*)d_in[3];
  const float* b2 = (const float*)d_in[4];
  const float* W3 = (const float*)d_in[5];
  const float* b3 = (const float*)d_in[6];
  float* out = (float*)d_out;
  float* x2  = (float*)d_ws;            // [16384, 512] f32 = 32 MB scratch

  dim3 blk(256);
  fc1_group_fused<<<dim3(B_SZ / 64, H1 / 64), blk, 0, stream>>>(x, W1, b1, W2,
                                                                b2, x2);
  fc3_kernel<<<dim3(B_SZ / 64, D_OUT / 64), blk, 0, stream>>>(x2, W3, b3, out);
}